// DiagTripleRCell_34402688041400
// MI455X (gfx1250) — compile-verified
//
#include <hip/hip_runtime.h>
#include <hip/hip_bf16.h>
#include <stdint.h>

typedef __attribute__((ext_vector_type(2))) float v2f;
typedef __attribute__((ext_vector_type(4))) float v4f;
typedef __attribute__((ext_vector_type(8))) float v8f;

#define T_DIM 2048
#define B_DIM 16
#define D_DIM 1024
#define M_DIM (T_DIM * B_DIM)   // 32768 rows of the projection GEMMs
#define K_DIM D_DIM
#define N_DIM D_DIM
#define S_DIM (B_DIM * D_DIM)   // 16384 elements per timestep slice

#define KC      64              // K-chunk staged per LDS buffer
#define NCHUNK  (K_DIM / KC)    // 16 chunks
#define NGRP    (KC / 8)        // 8 k-groups of 8 per chunk
#define APAD    68              // padded row stride (floats) in the A panel

#define WMMA_F32(acc, a, b) \
  __builtin_amdgcn_wmma_f32_16x16x4_f32(false, (a), false, (b), (short)0, (acc), false, false)

// -----------------------------------------------------------------------------
// Phase 1: out[m,n] = sum_k X[m,k] * W[n,k] + bias[n] for BOTH weight matrices.
// Block = 8 waves sharing one 16-row m-tile; wave w owns one 32-col n-group.
// A panel staged via global_load_async_to_lds_b128 (ASYNCcnt), double-buffered
// against compute, synced with s_wait_asynccnt + workgroup barrier.
//
// K-permutation: an 8-k group is split so wmma0 covers global k {0,1,4,5} and
// wmma1 covers {2,3,6,7} (lane-half h holds k = 4h+{0,1} / 4h+{2,3}).  The
// reduction is commutative, so the result is exact; the payoff is that each
// lane's A and B data for TWO wmmas is one contiguous b128:
//   4 global_load_b128 (B, 4 streams) + 1 ds_load_b128 (A)  per 8 wmmas.
// Loads for group g+1 are preloaded into distinct registers before group g's
// wmmas (software pipeline), with constant-offset addressing off 2 W bases.
// -----------------------------------------------------------------------------
__global__ __launch_bounds__(256)
void proj_gemm_kernel(const float* __restrict__ X,
                      const float* __restrict__ Wx,
                      const float* __restrict__ Wd,
                      const float* __restrict__ bias_x,
                      const float* __restrict__ bias_d,
                      float* __restrict__ outX,
                      float* __restrict__ outD) {
  __shared__ float smem[2][16 * APAD];

  const int tid  = threadIdx.x;
  const int wave = tid >> 5;
  const int lane = tid & 31;
  const int half = lane >> 4;    // 0: lanes 0-15, 1: lanes 16-31
  const int l16  = lane & 15;

  const int m_tile = blockIdx.x >> 2;                 // 2048 m-tiles
  const int n_grp  = ((blockIdx.x & 3) << 3) + wave;  // 32 n-groups of 32 cols
  const int m0 = m_tile << 4;
  const int n0 = n_grp << 5;

  // --- async staging slice for this thread: one b128 (16B) of the A panel ---
  const int arow = tid >> 4;                     // 0..15  (panel row)
  const int aseg = tid & 15;                     // 0..15  (16B segment in KC)
  const float* __restrict__ aglob = X + (size_t)(m0 + arow) * K_DIM + aseg * 4;

  // B stream bases (stream1 = base + 16 rows -> constant 65536B load offset)
  const float* __restrict__ wxb = Wx + (size_t)(n0 + l16) * K_DIM + 4 * half;
  const float* __restrict__ wdb = Wd + (size_t)(n0 + l16) * K_DIM + 4 * half;

  v8f ax0 = {}; v8f ax1 = {}; v8f ad0 = {}; v8f ad1 = {};

  // issue chunk 0 into buffer 0
  {
    unsigned lds_off = (unsigned)(uintptr_t)&smem[0][arow * APAD + aseg * 4];
    const float* g = aglob;
    asm volatile("global_load_async_to_lds_b128 %0, %1, off"
                 :: "v"(lds_off), "v"(g) : "memory");
  }

  for (int c = 0; c < NCHUNK; ++c) {
    // my async writes for chunk c are in LDS; barrier => everyone's are.
    asm volatile("s_wait_asynccnt 0x0" ::: "memory");
    __syncthreads();

    if (c + 1 < NCHUNK) {  // overlap next chunk's DMA with this chunk's math
      unsigned lds_off =
          (unsigned)(uintptr_t)&smem[(c + 1) & 1][arow * APAD + aseg * 4];
      const float* g = aglob + (c + 1) * KC;
      asm volatile("global_load_async_to_lds_b128 %0, %1, off"
                   :: "v"(lds_off), "v"(g) : "memory");
    }

    const float* __restrict__ ab = &smem[c & 1][l16 * APAD + 4 * half];
    const float* __restrict__ bx = wxb + c * KC;
    const float* __restrict__ bd = wdb + c * KC;

    // software pipeline: preload group 0, then overlap g+1 loads with g math
    v4f A  = *(const v4f*)(ab);
    v4f B0 = *(const v4f*)(bx);
    v4f B1 = *(const v4f*)(bx + 16 * K_DIM);
    v4f B2 = *(const v4f*)(bd);
    v4f B3 = *(const v4f*)(bd + 16 * K_DIM);

#pragma unroll
    for (int g = 0; g < NGRP; ++g) {
      v4f An = {}, B0n = {}, B1n = {}, B2n = {}, B3n = {};
      if (g + 1 < NGRP) {
        const int o = 8 * (g + 1);
        An  = *(const v4f*)(ab + o);
        B0n = *(const v4f*)(bx + o);
        B1n = *(const v4f*)(bx + o + 16 * K_DIM);
        B2n = *(const v4f*)(bd + o);
        B3n = *(const v4f*)(bd + o + 16 * K_DIM);
      }
      const v2f alo = __builtin_shufflevector(A, A, 0, 1);
      const v2f ahi = __builtin_shufflevector(A, A, 2, 3);
      ax0 = WMMA_F32(ax0, alo, __builtin_shufflevector(B0, B0, 0, 1));
      ax1 = WMMA_F32(ax1, alo, __builtin_shufflevector(B1, B1, 0, 1));
      ad0 = WMMA_F32(ad0, alo, __builtin_shufflevector(B2, B2, 0, 1));
      ad1 = WMMA_F32(ad1, alo, __builtin_shufflevector(B3, B3, 0, 1));
      ax0 = WMMA_F32(ax0, ahi, __builtin_shufflevector(B0, B0, 2, 3));
      ax1 = WMMA_F32(ax1, ahi, __builtin_shufflevector(B1, B1, 2, 3));
      ad0 = WMMA_F32(ad0, ahi, __builtin_shufflevector(B2, B2, 2, 3));
      ad1 = WMMA_F32(ad1, ahi, __builtin_shufflevector(B3, B3, 2, 3));
      A = An; B0 = B0n; B1 = B1n; B2 = B2n; B3 = B3n;
    }
  }

  const float bxc0 = bias_x[n0 + l16];
  const float bxc1 = bias_x[n0 + 16 + l16];
  const float bdc0 = bias_d[n0 + l16];
  const float bdc1 = bias_d[n0 + 16 + l16];

#pragma unroll
  for (int r = 0; r < 8; ++r) {
    const size_t row = (size_t)(m0 + r + 8 * half);
    outX[row * N_DIM + n0 + l16]      = ax0[r] + bxc0;
    outX[row * N_DIM + n0 + 16 + l16] = ax1[r] + bxc1;
    outD[row * N_DIM + n0 + l16]      = ad0[r] + bdc0;
    outD[row * N_DIM + n0 + 16 + l16] = ad1[r] + bdc1;
  }
}

// -----------------------------------------------------------------------------
// Phase 2: the recurrence. Fully parallel across (b,d): 16384 independent
// chains, one thread each, perfectly coalesced per-timestep slices.
// Software-pipelined loads (t+1 issued before t's transcendentals) plus
// global_prefetch 8 steps ahead to avoid latency serialization with only
// 512 waves in flight.
// -----------------------------------------------------------------------------
__global__ __launch_bounds__(256)
void scan_kernel(const float* __restrict__ x,
                 const float* __restrict__ h0,
                 const float* __restrict__ r_h,
                 const float* __restrict__ r_delta,
                 const float* __restrict__ b_gate,
                 const float* __restrict__ xW,
                 const float* __restrict__ xWd,
                 float* __restrict__ out,
                 float* __restrict__ hseq) {
  const int tid = blockIdx.x * blockDim.x + threadIdx.x;  // 0..16383
  const int d   = tid & (D_DIM - 1);
  const float rh = r_h[d];
  const float rd = r_delta[d];
  const float bg = b_gate[d];

  float h = h0[tid];
  hseq[tid] = h;  // h[0] = h0

  float xt  = x[tid];
  float xw  = xW[tid];
  float xwd = xWd[tid];

  for (int t = 0; t < T_DIM; ++t) {
    const size_t off = (size_t)t * S_DIM + tid;

    if (t + 8 < T_DIM) {
      const size_t poff = off + (size_t)8 * S_DIM;
      __builtin_prefetch(x + poff, 0, 1);
      __builtin_prefetch(xW + poff, 0, 1);
      __builtin_prefetch(xWd + poff, 0, 1);
    }

    float nx = 0.0f, nw = 0.0f, nwd = 0.0f;
    if (t + 1 < T_DIM) {  // issue next step's loads before this step's math
      nx  = x[off + S_DIM];
      nw  = xW[off + S_DIM];
      nwd = xWd[off + S_DIM];
    }

    const float v     = fmaf(rh, h, xw);
    const float cand  = tanhf(v);
    const float z     = fmaf(rd, h, xwd);
    const float delta = 1.0f / (1.0f + __expf(-z));
    const float hn    = fmaf(delta, cand - h, h);  // (1-delta)*h + delta*cand
    const float g     = hn + xt + bg;
    const float gate  = g / (1.0f + __expf(-g));   // silu(g)

    out[off]          = hn * gate;
    hseq[off + S_DIM] = hn;

    h = hn;
    xt = nx; xw = nw; xwd = nwd;
  }
}

extern "C" void kernel_launch(void* const* d_in, const int* in_sizes, int n_in,
                              void* d_out, int out_size, void* d_ws, size_t ws_size,
                              hipStream_t stream) {
  const float* x    = (const float*)d_in[0];
  const float* h0   = (const float*)d_in[1];
  const float* Wx   = (const float*)d_in[2];
  const float* Wd   = (const float*)d_in[3];
  const float* r_h  = (const float*)d_in[4];
  const float* r_d  = (const float*)d_in[5];
  const float* b    = (const float*)d_in[6];
  const float* b_d  = (const float*)d_in[7];
  const float* b_g  = (const float*)d_in[8];

  float* xW  = (float*)d_ws;                       // [T*B, D] fp32
  float* xWd = xW + (size_t)M_DIM * D_DIM;         // [T*B, D] fp32  (268 MB total)

  float* out  = (float*)d_out;                     // output [T,B,D]
  float* hseq = out + (size_t)T_DIM * S_DIM;       // h [T+1,B,D]

  // 2048 m-tiles * 4 n-parts = 8192 blocks; 8 waves/block -> 65536 wave-tiles.
  proj_gemm_kernel<<<8192, 256, 0, stream>>>(x, Wx, Wd, b, b_d, xW, xWd);
  // 16384 independent recurrence chains -> 64 blocks * 256 threads.
  scan_kernel<<<64, 256, 0, stream>>>(x, h0, r_h, r_d, b_g, xW, xWd, out, hseq);
}